// Attention_20667382629172
// MI455X (gfx1250) — compile-verified
//
#include <hip/hip_runtime.h>
#include <math.h>

// problem constants (match reference)
#define B_      64
#define H_      32
#define HKV_     8
#define G_       4          // H / HKV
#define D_     128
#define PAGE_   32
#define PPS_    64          // pages per sequence
#define SPLITS_  8
#define PGSPLIT_ 8          // pages per split
#define TOKSPLIT_ (PGSPLIT_*PAGE_)   // 256 tokens per split
#define SCALE_  0.08838834764831843f

typedef __attribute__((ext_vector_type(2))) float v2f;
typedef __attribute__((ext_vector_type(8))) float v8f;

// -------- kernel 1: flash-decode over one (b, kv, split) per wave ----------
__global__ __launch_bounds__(32)
void paged_attn_split(const float* __restrict__ q,
                      const float* __restrict__ knew,
                      const float* __restrict__ vnew,
                      const float* __restrict__ kcache,
                      const float* __restrict__ vcache,
                      const int* __restrict__ block_tables,
                      const int* __restrict__ context_lens,
                      const int* __restrict__ slot_mapping,
                      float* __restrict__ ws_m,
                      float* __restrict__ ws_l,
                      float* __restrict__ ws_acc)
{
    __shared__ float lds_q[16 * D_];            // Q group padded to 16 heads (rows 4..15 = 0)
    __shared__ float lds_kv[2][2 * PAGE_ * D_]; // double-buffered page: K(32x128) | V(32x128)
    __shared__ float lds_p[16 * 16];            // softmax probs, head-major

    const int lane  = threadIdx.x;
    const int wg    = blockIdx.x;
    const int split = wg % SPLITS_;
    const int kv    = (wg / SPLITS_) % HKV_;
    const int b     = wg / (SPLITS_ * HKV_);

    const int ctx      = context_lens[b];
    const int base_mls = ((b * HKV_ + kv) * SPLITS_ + split) * G_;

    if (split * TOKSPLIT_ >= ctx) {                 // empty split
        if (lane < G_) { ws_m[base_mls + lane] = -INFINITY; ws_l[base_mls + lane] = 0.f; }
        return;
    }
    const int tok_avail = ctx - split * TOKSPLIT_;
    const int npages    = (tok_avail + PAGE_ - 1) / PAGE_ < PGSPLIT_
                        ? (tok_avail + PAGE_ - 1) / PAGE_ : PGSPLIT_;

    // 64 scatter slots, striped across the wave (2 per lane)
    const int sm0 = slot_mapping[lane];
    const int sm1 = slot_mapping[32 + lane];

    // ---- stage Q group into LDS, zero-pad heads 4..15 ----
    for (int g = 0; g < G_; ++g) {
        const float4 qv = *(const float4*)(q + ((size_t)(b * H_ + kv * G_ + g)) * D_ + lane * 4);
        *(float4*)(lds_q + g * D_ + lane * 4) = qv;
    }
    const float4 z4 = make_float4(0.f, 0.f, 0.f, 0.f);
    for (int g = G_; g < 16; ++g)
        *(float4*)(lds_q + g * D_ + lane * 4) = z4;

    // issue one page of K/V as async global->LDS b128 copies (scatter-aware)
    auto issue_page = [&](int pp, int bufsel) {
        const int page_idx = split * PGSPLIT_ + pp;
        const int page     = block_tables[b * PPS_ + page_idx];
        float* kb = &lds_kv[bufsel][0];
        float* vb = &lds_kv[bufsel][PAGE_ * D_];
        for (int t = 0; t < PAGE_; ++t) {
            const int slot = page * PAGE_ + t;                 // wave-uniform
            const unsigned long long h0 = __ballot(sm0 == slot);
            const unsigned long long h1 = __ballot(sm1 == slot);
            const float *ksrc, *vsrc;
            if (h0 | h1) {                                     // new token overrides slot
                const int nbid = h0 ? (__ffsll(h0) - 1) : (32 + __ffsll(h1) - 1);
                ksrc = knew + ((size_t)nbid * HKV_ + kv) * D_;
                vsrc = vnew + ((size_t)nbid * HKV_ + kv) * D_;
            } else {
                ksrc = kcache + ((size_t)slot * HKV_ + kv) * D_;
                vsrc = vcache + ((size_t)slot * HKV_ + kv) * D_;
            }
            const unsigned ldsK = (unsigned)(size_t)(kb + t * D_ + lane * 4);
            const unsigned ldsV = (unsigned)(size_t)(vb + t * D_ + lane * 4);
            const unsigned long long gK = (unsigned long long)(size_t)(ksrc + lane * 4);
            const unsigned long long gV = (unsigned long long)(size_t)(vsrc + lane * 4);
            asm volatile("global_load_async_to_lds_b128 %0, %1, off"
                         :: "v"(ldsK), "v"(gK) : "memory");
            asm volatile("global_load_async_to_lds_b128 %0, %1, off"
                         :: "v"(ldsV), "v"(gV) : "memory");
        }
    };

    // flash state (per lane: head = lane&15)
    float m = -INFINITY, l = 0.f;
    v8f acc[8];
    for (int nc = 0; nc < 8; ++nc) acc[nc] = (v8f)0.0f;

    const int hr    = lane & 15;           // row/head index within half-wave
    const int hiOff = (lane >> 4) * 2;     // A/B fragment K-offset: 0 (lo) / 2 (hi)
    const int tHi   = (lane >> 4) * 8;     // token offset for score rows

    issue_page(0, 0);                      // prologue

    for (int pp = 0; pp < npages; ++pp) {
        const int tok_base = (split * PGSPLIT_ + pp) * PAGE_;

        // wait for page pp to land in LDS, then kick off pp+1 into other buffer
        asm volatile("s_wait_asynccnt 0" ::: "memory");
        if (pp + 1 < npages) {
            issue_page(pp + 1, (pp + 1) & 1);
            // warm L2 for page pp+2 (global_prefetch_b8)
            if (pp + 2 < npages) {
                const int np = block_tables[b * PPS_ + split * PGSPLIT_ + pp + 2];
                __builtin_prefetch(kcache + ((size_t)(np * PAGE_ + lane) * HKV_ + kv) * D_, 0, 0);
                __builtin_prefetch(vcache + ((size_t)(np * PAGE_ + lane) * HKV_ + kv) * D_, 0, 0);
            }
        }

        const float* kb = &lds_kv[pp & 1][0];
        const float* vb = &lds_kv[pp & 1][PAGE_ * D_];

        // ---- two 16-token subtiles per page ----
        for (int sub = 0; sub < 2; ++sub) {
            const int tok0 = tok_base + sub * 16;
            if (tok0 >= ctx) break;

            // scores^T[token, head]: 32 d-chunks of K=4, two interleaved accumulators
            v8f s0 = (v8f)0.0f, s1 = (v8f)0.0f;
            for (int kk = 0; kk < 32; kk += 2) {
                const int d0 = kk * 4 + hiOff;
                const v2f a0 = *(const v2f*)(kb + (sub * 16 + hr) * D_ + d0);
                const v2f b0 = *(const v2f*)(lds_q + hr * D_ + d0);
                const v2f a1 = *(const v2f*)(kb + (sub * 16 + hr) * D_ + d0 + 4);
                const v2f b1 = *(const v2f*)(lds_q + hr * D_ + d0 + 4);
                s0 = __builtin_amdgcn_wmma_f32_16x16x4_f32(false, a0, false, b0,
                                                           (short)0, s0, false, false);
                s1 = __builtin_amdgcn_wmma_f32_16x16x4_f32(false, a1, false, b1,
                                                           (short)0, s1, false, false);
            }
            const v8f s = s0 + s1;

            // ---- flash softmax update; lane's head=hr, tokens r+tHi ----
            float val[8]; float tile_m = -INFINITY;
            for (int r = 0; r < 8; ++r) {
                const int tok = tok0 + r + tHi;
                val[r] = (tok < ctx) ? s[r] * SCALE_ : -INFINITY;
                tile_m = fmaxf(tile_m, val[r]);
            }
            tile_m = fmaxf(tile_m, __shfl_xor(tile_m, 16, 32));
            const float m_new = fmaxf(m, tile_m);       // finite: >=1 valid token
            float pv[8]; float tile_l = 0.f;
            for (int r = 0; r < 8; ++r) {
                const float e = (val[r] == -INFINITY) ? 0.f : __expf(val[r] - m_new);
                pv[r] = e; tile_l += e;
            }
            tile_l += __shfl_xor(tile_l, 16, 32);
            const float sc_prev = __expf(m - m_new);    // exp(-inf - finite) = 0 first time
            l = l * sc_prev + tile_l;
            m = m_new;

            // rescale accumulators: acc row r holds head r (lo lanes) / r+8 (hi lanes)
            for (int r = 0; r < 8; ++r) {
                const float slo = __shfl(sc_prev, r, 32);
                const float shi = __shfl(sc_prev, r + 8, 32);
                const float sr  = (lane < 16) ? slo : shi;
                for (int nc = 0; nc < 8; ++nc) acc[nc][r] *= sr;
            }

            // park P in LDS head-major (lane h -> tokens 0..7, lane h+16 -> 8..15)
            *(float4*)(lds_p + hr * 16 + tHi + 0) = make_float4(pv[0], pv[1], pv[2], pv[3]);
            *(float4*)(lds_p + hr * 16 + tHi + 4) = make_float4(pv[4], pv[5], pv[6], pv[7]);

            // ---- out += P x V : M=16 heads, K=4 tokens, N=16 d (8 indep chains) ----
            for (int tk = 0; tk < 4; ++tk) {
                const v2f ap   = *(const v2f*)(lds_p + hr * 16 + tk * 4 + hiOff);
                const int row0 = sub * 16 + tk * 4 + hiOff;
                for (int nc = 0; nc < 8; ++nc) {
                    v2f bv;
                    bv.x = vb[(row0    ) * D_ + nc * 16 + hr];
                    bv.y = vb[(row0 + 1) * D_ + nc * 16 + hr];
                    acc[nc] = __builtin_amdgcn_wmma_f32_16x16x4_f32(false, ap, false, bv,
                                                                    (short)0, acc[nc], false, false);
                }
            }
        }
    }

    // ---- write split partials ----
    if (lane < G_) { ws_m[base_mls + lane] = m; ws_l[base_mls + lane] = l; }
    if (lane < 16) {
        for (int g = 0; g < G_; ++g)
            for (int nc = 0; nc < 8; ++nc)
                ws_acc[(size_t)(base_mls + g) * D_ + nc * 16 + lane] = acc[nc][g];
    }
}

// -------- kernel 2: log-sum-exp combine over splits ----------
__global__ __launch_bounds__(128)
void paged_attn_combine(const float* __restrict__ ws_m,
                        const float* __restrict__ ws_l,
                        const float* __restrict__ ws_acc,
                        float* __restrict__ out)
{
    const int bkv = blockIdx.x;       // b*HKV + kv
    const int d   = threadIdx.x;      // 0..127
    for (int g = 0; g < G_; ++g) {
        float ms[SPLITS_]; float M = -INFINITY;
        for (int s = 0; s < SPLITS_; ++s) {
            ms[s] = ws_m[(bkv * SPLITS_ + s) * G_ + g];
            M = fmaxf(M, ms[s]);
        }
        float num = 0.f, den = 0.f;
        for (int s = 0; s < SPLITS_; ++s) {
            const float sc = (ms[s] == -INFINITY) ? 0.f : __expf(ms[s] - M);
            if (sc != 0.f) {
                den += sc * ws_l[(bkv * SPLITS_ + s) * G_ + g];
                num += sc * ws_acc[(size_t)((bkv * SPLITS_ + s) * G_ + g) * D_ + d];
            }
        }
        out[(size_t)(bkv * G_ + g) * D_ + d] = num / den;
    }
}

extern "C" void kernel_launch(void* const* d_in, const int* in_sizes, int n_in,
                              void* d_out, int out_size, void* d_ws, size_t ws_size,
                              hipStream_t stream) {
    (void)in_sizes; (void)n_in; (void)out_size; (void)ws_size;
    const float* q        = (const float*)d_in[0];
    const float* knew     = (const float*)d_in[1];
    const float* vnew     = (const float*)d_in[2];
    const float* kcache   = (const float*)d_in[3];
    const float* vcache   = (const float*)d_in[4];
    const int*   btables  = (const int*)d_in[5];
    const int*   ctxlens  = (const int*)d_in[6];
    const int*   slots    = (const int*)d_in[7];
    float*       out      = (float*)d_out;

    const size_t MLS = (size_t)B_ * HKV_ * SPLITS_ * G_;   // 16384
    float* ws_m   = (float*)d_ws;
    float* ws_l   = ws_m + MLS;
    float* ws_acc = ws_l + MLS;                            // MLS * 128 floats (~8 MB)

    paged_attn_split<<<B_ * HKV_ * SPLITS_, 32, 0, stream>>>(
        q, knew, vnew, kcache, vcache, btables, ctxlens, slots, ws_m, ws_l, ws_acc);
    paged_attn_combine<<<B_ * HKV_, 128, 0, stream>>>(ws_m, ws_l, ws_acc, out);
}